// MoE_592705487376
// MI455X (gfx1250) — compile-verified
//
#include <hip/hip_runtime.h>
#include <math.h>

// MoE fused FFN for MI455X (gfx1250), FP32 via V_WMMA_F32_16X16X4_F32.
// B=2,S=2048 -> T=4096 tokens, H=1024, F=512, E=8, top-2.
// Gathered (sparse) execution, 32-token tiles: weight L2 traffic ~1.6 GB
// (2x less than 16-token tiles); every B fragment feeds 2 (phase1) / 2
// (phase2) WMMAs. acts buffer aliases the x tile in LDS (131.8 KB total).

typedef float v2f __attribute__((ext_vector_type(2)));
typedef float v8f __attribute__((ext_vector_type(8)));

#define T_TOKENS 4096
#define H_DIM    1024
#define F_DIM    512
#define E_NUM    8
#define TILE_M   32
#define XSTR     (H_DIM + 4)   // 1028 dwords; 1028%64==4 -> conflict-free LDS
#define ASTR     (F_DIM + 4)   // 516 dwords;   516%64==4 -> conflict-free LDS

#define WMMA_F32(A, B, C) \
  __builtin_amdgcn_wmma_f32_16x16x4_f32(false, (A), false, (B), (short)0, (C), false, false)

// ---------------------------------------------------------------------------
// Router: one wave32 per token. logits = x[t,:] @ Wr, softmax, top-2,
// renormalize, scatter (token, weight) into per-expert lists.
// ---------------------------------------------------------------------------
__global__ __launch_bounds__(128) void moe_router_kernel(
    const float* __restrict__ x, const float* __restrict__ Wr,
    int* __restrict__ counts, int* __restrict__ tok, float* __restrict__ wt)
{
  const int lane = threadIdx.x & 31;
  const int wave = threadIdx.x >> 5;
  const int t = blockIdx.x * 4 + wave;
  if (t >= T_TOKENS) return;

  float acc[E_NUM];
#pragma unroll
  for (int e = 0; e < E_NUM; ++e) acc[e] = 0.0f;

  const float* xrow = x + (size_t)t * H_DIM;
  for (int h = lane; h < H_DIM; h += 32) {
    float xv = xrow[h];
    const float4* wr = (const float4*)(Wr + (size_t)h * E_NUM);
    float4 w0 = wr[0];
    float4 w1 = wr[1];
    acc[0] = fmaf(xv, w0.x, acc[0]);
    acc[1] = fmaf(xv, w0.y, acc[1]);
    acc[2] = fmaf(xv, w0.z, acc[2]);
    acc[3] = fmaf(xv, w0.w, acc[3]);
    acc[4] = fmaf(xv, w1.x, acc[4]);
    acc[5] = fmaf(xv, w1.y, acc[5]);
    acc[6] = fmaf(xv, w1.z, acc[6]);
    acc[7] = fmaf(xv, w1.w, acc[7]);
  }
#pragma unroll
  for (int e = 0; e < E_NUM; ++e) {
#pragma unroll
    for (int off = 16; off > 0; off >>= 1)
      acc[e] += __shfl_xor(acc[e], off, 32);
  }
  float mx = acc[0];
#pragma unroll
  for (int e = 1; e < E_NUM; ++e) mx = fmaxf(mx, acc[e]);
  float p[E_NUM];
  float s = 0.0f;
#pragma unroll
  for (int e = 0; e < E_NUM; ++e) { p[e] = expf(acc[e] - mx); s += p[e]; }
  float inv_s = 1.0f / s;
#pragma unroll
  for (int e = 0; e < E_NUM; ++e) p[e] *= inv_s;
  int i1 = 0; float v1 = p[0];
#pragma unroll
  for (int e = 1; e < E_NUM; ++e) if (p[e] > v1) { v1 = p[e]; i1 = e; }
  int i2 = (i1 == 0) ? 1 : 0; float v2 = p[i2];
#pragma unroll
  for (int e = 0; e < E_NUM; ++e)
    if (e != i1 && p[e] > v2) { v2 = p[e]; i2 = e; }
  float inv12 = 1.0f / (v1 + v2);

  if (lane == 0) {
    int p1 = atomicAdd(&counts[i1], 1);
    tok[(size_t)i1 * T_TOKENS + p1] = t;
    wt [(size_t)i1 * T_TOKENS + p1] = v1 * inv12;
    int p2 = atomicAdd(&counts[i2], 1);
    tok[(size_t)i2 * T_TOKENS + p2] = t;
    wt [(size_t)i2 * T_TOKENS + p2] = v2 * inv12;
  }
}

// ---------------------------------------------------------------------------
// Expert FFN: one block = 32 gathered tokens of expert e; 8 waves.
// Phase 1 (gate/up): wave owns 64 F-cols; k outer, 4 n-tiles x 2 row-blocks
//   inner (16 accumulator tiles, 128 VGPRs); each Wg/Wu fragment feeds 2
//   WMMAs (row-block reuse).  After one barrier, silu(g)*u*route_w is
//   written into LDS, aliasing the now-dead x tile.
// Phase 2 (down): wave owns 128 H-cols; k outer, 8 n-tiles x 2 row-blocks
//   (128 VGPR accum); each Wd fragment feeds 2 WMMAs.  Scatter-accumulate
//   to out with atomicAdd (exactly 2 addends per element -> deterministic).
// ---------------------------------------------------------------------------
__global__ __launch_bounds__(256) void moe_expert_ffn_kernel(
    const float* __restrict__ x,
    const int*   __restrict__ counts,
    const int*   __restrict__ tok,
    const float* __restrict__ wt,
    const float* __restrict__ Wg,
    const float* __restrict__ Wu,
    const float* __restrict__ Wd,
    float* __restrict__ out)
{
  // 32*1028*4 = 131,584 B; phase-2 'acts' [32][516] aliases the same buffer.
  __shared__ __align__(16) float smem[TILE_M * XSTR];
  __shared__ int   ts[TILE_M];
  __shared__ float rw[TILE_M];

  const int e    = blockIdx.y;
  const int tile = blockIdx.x;
  const int cnt  = counts[e];
  if (tile * TILE_M >= cnt) return;            // uniform early-out

  const int tid  = threadIdx.x;
  const int lane = tid & 31;
  const int wave = tid >> 5;
  const int lm   = lane & 15;        // A row / B,C,D column within 16-tile
  const int kg   = (lane >> 4) << 1; // K sub-block base: 0 (lanes 0-15) or 2
  const int mhi  = (lane < 16) ? 0 : 8;

  if (tid < TILE_M) {
    int idx = tile * TILE_M + tid;
    if (idx < cnt) {
      ts[tid] = tok[(size_t)e * T_TOKENS + idx];
      rw[tid] = wt [(size_t)e * T_TOKENS + idx];
    } else {
      ts[tid] = -1;
      rw[tid] = 0.0f;
    }
  }
  __syncthreads();

  // stage gathered x rows into LDS (float4; row base m*4112 B, 16B aligned)
  for (int i = tid; i < TILE_M * (H_DIM / 4); i += 256) {
    int m = i >> 8;          // H_DIM/4 == 256
    int c = i & 255;
    int tkn = ts[m];
    float4 v = make_float4(0.f, 0.f, 0.f, 0.f);
    if (tkn >= 0) v = ((const float4*)(x + (size_t)tkn * H_DIM))[c];
    ((float4*)&smem[m * XSTR])[c] = v;
  }
  __syncthreads();

  const int f_base = wave * 64;     // phase-1 F chunk
  const int h_base = wave * 128;    // phase-2 H chunk

  // ---------------- Phase 1: gate/up, K = H = 1024 ----------------
  v8f accg[2][4], accu[2][4];
#pragma unroll
  for (int rb = 0; rb < 2; ++rb)
#pragma unroll
    for (int nt = 0; nt < 4; ++nt) {
      v8f z = {0.f,0.f,0.f,0.f,0.f,0.f,0.f,0.f};
      accg[rb][nt] = z;
      accu[rb][nt] = z;
    }
  {
    const float* pg = Wg + (size_t)e * H_DIM * F_DIM + (size_t)kg * F_DIM + f_base + lm;
    const float* pu = Wu + (size_t)e * H_DIM * F_DIM + (size_t)kg * F_DIM + f_base + lm;
    const float* a0p = &smem[lm * XSTR + kg];
    const float* a1p = &smem[(lm + 16) * XSTR + kg];
    for (int k = 0; k < H_DIM; k += 4) {
      v2f a0; a0.x = a0p[k]; a0.y = a0p[k + 1];
      v2f a1; a1.x = a1p[k]; a1.y = a1p[k + 1];
#pragma unroll
      for (int nt = 0; nt < 4; ++nt) {
        v2f bg; bg.x = pg[nt * 16]; bg.y = pg[nt * 16 + F_DIM];
        v2f bu; bu.x = pu[nt * 16]; bu.y = pu[nt * 16 + F_DIM];
        accg[0][nt] = WMMA_F32(a0, bg, accg[0][nt]);
        accg[1][nt] = WMMA_F32(a1, bg, accg[1][nt]);
        accu[0][nt] = WMMA_F32(a0, bu, accu[0][nt]);
        accu[1][nt] = WMMA_F32(a1, bu, accu[1][nt]);
      }
      pg += 4 * F_DIM;
      pu += 4 * F_DIM;
    }
  }
  __syncthreads();   // all waves done reading the x tile

  // act = silu(gate)*up*route_w  ->  LDS (aliased over x tile)
#pragma unroll
  for (int rb = 0; rb < 2; ++rb) {
#pragma unroll
    for (int nt = 0; nt < 4; ++nt) {
#pragma unroll
      for (int j = 0; j < 8; ++j) {
        int m = rb * 16 + mhi + j;   // D layout: VGPR j -> M=j / j+8
        float gv = accg[rb][nt][j];
        float uv = accu[rb][nt][j];
        float sv = gv / (1.0f + __expf(-gv));
        smem[m * ASTR + f_base + nt * 16 + lm] = sv * uv * rw[m];
      }
    }
  }
  __syncthreads();

  // ---------------- Phase 2: down-proj, K = F = 512 ----------------
  v8f dacc[2][8];
#pragma unroll
  for (int rb = 0; rb < 2; ++rb)
#pragma unroll
    for (int nt = 0; nt < 8; ++nt) {
      v8f z = {0.f,0.f,0.f,0.f,0.f,0.f,0.f,0.f};
      dacc[rb][nt] = z;
    }
  {
    const float* pd = Wd + (size_t)e * F_DIM * H_DIM + (size_t)kg * H_DIM + h_base + lm;
    const float* a0p = &smem[lm * ASTR + kg];
    const float* a1p = &smem[(lm + 16) * ASTR + kg];
    for (int k = 0; k < F_DIM; k += 4) {
      v2f a0; a0.x = a0p[k]; a0.y = a0p[k + 1];
      v2f a1; a1.x = a1p[k]; a1.y = a1p[k + 1];
#pragma unroll
      for (int nt = 0; nt < 8; ++nt) {
        v2f b; b.x = pd[nt * 16]; b.y = pd[nt * 16 + H_DIM];
        dacc[0][nt] = WMMA_F32(a0, b, dacc[0][nt]);
        dacc[1][nt] = WMMA_F32(a1, b, dacc[1][nt]);
      }
      pd += 4 * H_DIM;
    }
  }

  // Each token receives exactly 2 expert contributions (top-2); 2-addend
  // float atomic adds are commutative bitwise -> deterministic output.
#pragma unroll
  for (int rb = 0; rb < 2; ++rb) {
#pragma unroll
    for (int nt = 0; nt < 8; ++nt) {
#pragma unroll
      for (int j = 0; j < 8; ++j) {
        int m = rb * 16 + mhi + j;
        int tkn = ts[m];
        if (tkn >= 0)
          atomicAdd(out + (size_t)tkn * H_DIM + h_base + nt * 16 + lm,
                    dacc[rb][nt][j]);
      }
    }
  }
}

// ---------------------------------------------------------------------------
// Launch. Workspace layout: [counts: 8 int (zeroed, 64B slot)]
//                           [tok: 8*4096 int] [wt: 8*4096 float]  (~256 KB)
// ---------------------------------------------------------------------------
extern "C" void kernel_launch(void* const* d_in, const int* in_sizes, int n_in,
                              void* d_out, int out_size, void* d_ws, size_t ws_size,
                              hipStream_t stream) {
  const float* x  = (const float*)d_in[0];
  const float* Wr = (const float*)d_in[1];
  const float* Wg = (const float*)d_in[2];
  const float* Wu = (const float*)d_in[3];
  const float* Wd = (const float*)d_in[4];
  // d_in[5] = topk (==2), hardcoded in the kernels.
  float* out = (float*)d_out;

  int*   counts = (int*)d_ws;
  int*   tok    = (int*)((char*)d_ws + 64);
  float* wtp    = (float*)((char*)d_ws + 64 + (size_t)E_NUM * T_TOKENS * sizeof(int));

  hipMemsetAsync(d_ws, 0, 64, stream);                                  // counts
  hipMemsetAsync(d_out, 0, (size_t)out_size * sizeof(float), stream);   // accum target

  moe_router_kernel<<<T_TOKENS / 4, 128, 0, stream>>>(x, Wr, counts, tok, wtp);

  dim3 grid(T_TOKENS / TILE_M, E_NUM);   // 128 x 8; inactive tiles early-out
  moe_expert_ffn_kernel<<<grid, 256, 0, stream>>>(x, counts, tok, wtp,
                                                  Wg, Wu, Wd, out);
}